// InverseAvgPool1d_48850958025252
// MI455X (gfx1250) — compile-verified
//
#include <hip/hip_runtime.h>
#include <stdint.h>

// InverseAvgPool1d (K=8): out[t] = out[t-9] + 8*(x[t]-x[t-1]), d[0]=0, +8*x[0] at t=5.
// One workgroup per (b,c) row of 4096 f32. 9 waves: wave r owns residue chain t ≡ r (mod 9).
// Bandwidth-bound (512 MB @ 23.3 TB/s ≈ 22 us floor) -> async global<->LDS staging + a
// work-efficient two-pass per-chain scan (lane-serial chunk + wave32 shuffle scan).

#define TLEN  4096
#define NRES  9
#define BLOCK 288          // 9 waves * 32 lanes
#define CHUNK 15           // ceil(456 / 32)
#define KP    8.0f
#define CPOL_NT 1          // TH[2:0] = NT: one-shot stream, don't thrash 192MB L2

// Builtin signature (from clang diagnostic): params are v4i* in AS1 (global) / AS3 (LDS).
typedef int v4i __attribute__((vector_size(16)));
typedef __attribute__((address_space(1))) v4i gv4i;
typedef __attribute__((address_space(3))) v4i lv4i;
// Addrspace cast via integer round-trip (flat->LDS keeps offset in addr[31:0] per ISA).
#define GCAST(p) ((gv4i*)(uintptr_t)(p))
#define LCAST(p) ((lv4i*)(unsigned)(uintptr_t)(p))

#if defined(__gfx1250__) && \
    __has_builtin(__builtin_amdgcn_global_load_async_to_lds_b128) && \
    __has_builtin(__builtin_amdgcn_global_store_async_from_lds_b128)
#define USE_ASYNC_LDS 1
#else
#define USE_ASYNC_LDS 0
#endif

__global__ __launch_bounds__(BLOCK)
void inverse_avgpool1d_kernel(const float* __restrict__ x, float* __restrict__ out) {
  __shared__ __align__(16) float xs[TLEN];
  __shared__ __align__(16) float os[TLEN];

  const int    tid  = threadIdx.x;
  const size_t base = (size_t)blockIdx.x * TLEN;

  // ---- Stage one 16 KB row into LDS (coalesced, B128 per lane, ASYNCcnt-tracked) ----
#if USE_ASYNC_LDS
  for (int i = tid; i < TLEN / 4; i += BLOCK) {
    __builtin_amdgcn_global_load_async_to_lds_b128(
        GCAST(x + base + 4 * i), LCAST(&xs[4 * i]), 0, CPOL_NT);
  }
#if __has_builtin(__builtin_amdgcn_s_wait_asynccnt)
  __builtin_amdgcn_s_wait_asynccnt(0);
#else
  asm volatile("s_wait_asynccnt 0" ::: "memory");
#endif
#else
  for (int i = tid; i < TLEN / 4; i += BLOCK) {
    ((float4*)xs)[i] = ((const float4*)(x + base))[i];
  }
#endif
  __syncthreads();

  // ---- Per-chain scan: wave r handles t = r, r+9, r+18, ... ----
  const int wave = tid >> 5;        // residue r in [0, 9)
  const int lane = tid & 31;
  const int r    = wave;
  const int Lr   = (TLEN - 1 - r) / NRES + 1;     // chain length (455 or 456)
  const int m0   = lane * CHUNK;
  const int mEnd = (m0 + CHUNK < Lr) ? (m0 + CHUNK) : Lr;

  // Pass 1: lane-local sum of d over its chunk (LDS stride-9 reads: bank-conflict-free,
  // addresses hit banks 7*lane mod 64 across the wave).
  float s = 0.0f;
  for (int m = m0; m < mEnd; ++m) {
    const int t = NRES * m + r;
    float d = (t == 0) ? 0.0f : KP * (xs[t] - xs[t - 1]);
    if (t == 5) d += KP * xs[0];
    s += d;
  }

  // Wave32 inclusive shuffle-scan of lane sums -> exclusive offset per lane.
  float v = s;
  #pragma unroll
  for (int dlt = 1; dlt < 32; dlt <<= 1) {
    float u = __shfl_up(v, dlt, 32);
    if (lane >= dlt) v += u;
  }
  const float excl = v - s;

  // Pass 2: emit running sums into the output LDS buffer.
  float run = excl;
  for (int m = m0; m < mEnd; ++m) {
    const int t = NRES * m + r;
    float d = (t == 0) ? 0.0f : KP * (xs[t] - xs[t - 1]);
    if (t == 5) d += KP * xs[0];
    run += d;
    os[t] = run;
  }
  __syncthreads();

  // ---- Drain LDS row back to global (coalesced, B128 per lane) ----
#if USE_ASYNC_LDS
  for (int i = tid; i < TLEN / 4; i += BLOCK) {
    __builtin_amdgcn_global_store_async_from_lds_b128(
        GCAST(out + base + 4 * i), LCAST(&os[4 * i]), 0, CPOL_NT);
  }
  // S_ENDPGM performs an implicit wait-idle; ASYNCcnt drains before the wave retires.
#else
  for (int i = tid; i < TLEN / 4; i += BLOCK) {
    ((float4*)(out + base))[i] = ((const float4*)os)[i];
  }
#endif
}

extern "C" void kernel_launch(void* const* d_in, const int* in_sizes, int n_in,
                              void* d_out, int out_size, void* d_ws, size_t ws_size,
                              hipStream_t stream) {
  (void)in_sizes; (void)n_in; (void)d_ws; (void)ws_size;
  const float* x = (const float*)d_in[0];
  float* out = (float*)d_out;
  const int rows = out_size / TLEN;   // 64 * 256 = 16384
  inverse_avgpool1d_kernel<<<dim3(rows), dim3(BLOCK), 0, stream>>>(x, out);
}